// ContTimeLSTMCell_1133871366520
// MI455X (gfx1250) — compile-verified
//
#include <hip/hip_runtime.h>

typedef float v2f __attribute__((ext_vector_type(2)));
typedef float v8f __attribute__((ext_vector_type(8)));

constexpr int EMBED  = 256;
constexpr int HIDDEN = 1024;
constexpr int BATCH  = 8192;
constexpr int FANIN  = HIDDEN + EMBED;   // 1280
constexpr int NG     = 7;

constexpr int BT = 64;        // batch tile per workgroup
constexpr int HT = 32;        // hidden tile per workgroup
constexpr int KC = 32;        // K chunk staged in LDS
constexpr int LDX = KC + 2;   // 34 floats: row stride (8B-aligned) for X tile
constexpr int LDW = HT + 1;   // 33 floats: row stride for W tiles

__global__ __launch_bounds__(256)
void ctlstm_fused(const int*   __restrict__ evt,
                  const float* __restrict__ hidden,
                  const float* __restrict__ cell_prev,
                  const float* __restrict__ cellbar_prev,
                  const float* __restrict__ emb,
                  const float* __restrict__ W,
                  const float* __restrict__ bias,
                  float*       __restrict__ out)
{
    __shared__ float sX[BT][LDX];          // X tile:   64 x 32 (+pad)
    __shared__ float sW[NG][KC][LDW];      // W tiles: 7 x 32 x 32 (+pad)

    const int t    = threadIdx.x;
    const int lane = t & 31;
    const int wv   = t >> 5;          // 0..7
    const int bw   = wv & 3;          // 4 waves along B
    const int hw   = wv >> 2;         // 2 waves along H
    const int bBase = blockIdx.x * BT;
    const int hBase = blockIdx.y * HT;

    // --- cooperative loader coordinates ---
    const int xr = t >> 2;            // 0..63 : X row
    const int xc = (t & 3) * 8;       // 8 floats per thread per X row
    const int wr = t >> 3;            // 0..31 : W row (k within chunk)
    const int wc = (t & 7) * 4;       // 4 floats per thread per W row (per gate)

    const int    evrow  = evt[bBase + xr];
    const float* embRow = emb    + (size_t)evrow * EMBED;
    const float* hidRow = hidden + (size_t)(bBase + xr) * HIDDEN;

    // --- WMMA fragment coordinates (ISA 7.12.2, f32 16x16x4) ---
    // A: lanes 0-15 hold M=lane, VGPR0/1 = K,K+1; lanes 16-31 = K+2,K+3
    const int aRow = bw * 16 + (lane & 15);
    const int kSel = (lane >> 4) * 2;            // 0 or 2
    // B: row K striped across lanes; mirror of A layout
    const int nCol = hw * 16 + (lane & 15);

    v8f acc[NG] = {};

    for (int k0 = 0; k0 < FANIN; k0 += KC) {
        // ---- stage X tile (emb-gather fused for k < EMBED) ----
        {
            const float* src = (k0 < EMBED) ? (embRow + k0 + xc)
                                            : (hidRow + (k0 - EMBED) + xc);
            const float4 v0 = *(const float4*)(src);
            const float4 v1 = *(const float4*)(src + 4);
            sX[xr][xc + 0] = v0.x; sX[xr][xc + 1] = v0.y;
            sX[xr][xc + 2] = v0.z; sX[xr][xc + 3] = v0.w;
            sX[xr][xc + 4] = v1.x; sX[xr][xc + 5] = v1.y;
            sX[xr][xc + 6] = v1.z; sX[xr][xc + 7] = v1.w;
        }
        // ---- stage W tiles: 7 gates x [KC x HT] ----
        #pragma unroll
        for (int g = 0; g < NG; ++g) {
            const float4 v = *(const float4*)(
                W + ((size_t)g * FANIN + (k0 + wr)) * HIDDEN + hBase + wc);
            sW[g][wr][wc + 0] = v.x; sW[g][wr][wc + 1] = v.y;
            sW[g][wr][wc + 2] = v.z; sW[g][wr][wc + 3] = v.w;
        }
        __syncthreads();

        // ---- 8 k-steps x 7 gates of V_WMMA_F32_16X16X4_F32 ----
        #pragma unroll
        for (int kk = 0; kk < KC; kk += 4) {
            v2f a;
            a.x = sX[aRow][kk + kSel];
            a.y = sX[aRow][kk + kSel + 1];
            #pragma unroll
            for (int g = 0; g < NG; ++g) {
                v2f bf;
                bf.x = sW[g][kk + kSel][nCol];
                bf.y = sW[g][kk + kSel + 1][nCol];
                acc[g] = __builtin_amdgcn_wmma_f32_16x16x4_f32(
                    false, a, false, bf, (short)0, acc[g], false, false);
            }
        }
        __syncthreads();
    }

    // ---- epilogue: bias + gates + cell updates ----
    const int n    = lane & 15;
    const int mHi  = (lane >> 4) * 8;           // C/D: VGPR r -> rows r, r+8
    const int hOut = hBase + hw * 16 + n;

    float bv[NG];
    #pragma unroll
    for (int g = 0; g < NG; ++g) bv[g] = bias[g * HIDDEN + hOut];

    const size_t BH = (size_t)BATCH * HIDDEN;
    float* outCell    = out;
    float* outCellBar = out + BH;
    float* outDecay   = out + 2 * BH;
    float* outGate    = out + 3 * BH;

    #pragma unroll
    for (int r = 0; r < 8; ++r) {
        const int    b   = bBase + bw * 16 + mHi + r;
        const size_t idx = (size_t)b * HIDDEN + hOut;

        const float z0 = acc[0][r] + bv[0];   // input
        const float z1 = acc[1][r] + bv[1];   // forget
        const float z2 = acc[2][r] + bv[2];   // output
        const float z3 = acc[3][r] + bv[3];   // input_bar
        const float z4 = acc[4][r] + bv[4];   // forget_bar
        const float z5 = acc[5][r] + bv[5];   // pre_c
        const float z6 = acc[6][r] + bv[6];   // decay

        const float gi  = 1.0f / (1.0f + expf(-z0));
        const float gf  = 1.0f / (1.0f + expf(-z1));
        const float go  = 1.0f / (1.0f + expf(-z2));
        const float gib = 1.0f / (1.0f + expf(-z3));
        const float gfb = 1.0f / (1.0f + expf(-z4));
        const float pc  = tanhf(z5);
        const float sp  = fmaxf(z6, 0.0f) + log1pf(expf(-fabsf(z6)));

        const float cprev  = cell_prev[idx];
        const float cbprev = cellbar_prev[idx];

        outCell[idx]    = gf  * cprev  + gi  * pc;
        outCellBar[idx] = gfb * cbprev + gib * pc;
        outDecay[idx]   = sp;
        outGate[idx]    = go;
    }
}

extern "C" void kernel_launch(void* const* d_in, const int* in_sizes, int n_in,
                              void* d_out, int out_size, void* d_ws, size_t ws_size,
                              hipStream_t stream) {
    const int*   evt     = (const int*)  d_in[0];
    const float* hidden  = (const float*)d_in[1];
    const float* cell    = (const float*)d_in[2];
    const float* cellbar = (const float*)d_in[3];
    const float* emb     = (const float*)d_in[4];
    const float* W       = (const float*)d_in[5];
    const float* b       = (const float*)d_in[6];

    dim3 grid(BATCH / BT, HIDDEN / HT);
    ctlstm_fused<<<grid, 256, 0, stream>>>(evt, hidden, cell, cellbar,
                                           emb, W, b, (float*)d_out);
}